// SeqSelfAttention_7851200217353
// MI455X (gfx1250) — compile-verified
//
#include <hip/hip_runtime.h>
#include <hip/hip_bf16.h>
#include <math.h>

// Reference shapes (fixed by setup_inputs): B=4, T=1024, F=256, U=32, band=64.
#define BATCH 4
#define TLEN  1024
#define FDIM  256
#define UDIM  32
#define BAND  64
#define SWIN  80   // union of bands over a 16-row t-tile: [t0-32, t0+47] -> 80 keys

typedef __attribute__((ext_vector_type(2))) float v2f;
typedef __attribute__((ext_vector_type(8))) float v8f;

// ---------------------------------------------------------------------------
// Kernel 1: q = x @ Wt ; k = x @ Wx + bh      (rows = B*T = 4096, K = 256)
// One wave (32 threads) per 16x16 output tile using V_WMMA_F32_16X16X4_F32.
// grid.x = 4096/16 = 256 row tiles; grid.y = 4 (part 0,1 -> q cols; 2,3 -> k cols)
// ---------------------------------------------------------------------------
__global__ __launch_bounds__(32) void proj_wmma_kernel(
    const float* __restrict__ x,   // [B*T, F]
    const float* __restrict__ Wt,  // [F, U]
    const float* __restrict__ Wx,  // [F, U]
    const float* __restrict__ bh,  // [U]
    float* __restrict__ qo,        // [B*T, U]
    float* __restrict__ ko)        // [B*T, U]
{
    const int lane    = threadIdx.x;            // 0..31
    const int rowBase = blockIdx.x * 16;
    const int part    = blockIdx.y;             // 0..3
    const bool isK    = (part >= 2);
    const int colBase = (part & 1) * 16;
    const float* __restrict__ W = isK ? Wx : Wt;
    float* __restrict__ out     = isK ? ko : qo;

    const int mrow  = lane & 15;                // A row for this lane
    const int khalf = (lane >> 4) * 2;          // lanes 0-15 -> K0,K1 ; 16-31 -> K2,K3
    const int n     = lane & 15;                // B/D column for this lane

    const float* __restrict__ xrow = x + (size_t)(rowBase + mrow) * FDIM + khalf;

    v8f c = {};
    for (int k0 = 0; k0 < FDIM; k0 += 4) {
        // A fragment (16x4 f32): lane holds A[mrow][k0+khalf .. +1] (8B-aligned)
        v2f a = *(const v2f*)(xrow + k0);
        // B fragment (4x16 f32): lane holds W[k0+khalf][colBase+n], W[k0+khalf+1][...]
        v2f b;
        b.x = W[(k0 + khalf    ) * UDIM + colBase + n];
        b.y = W[(k0 + khalf + 1) * UDIM + colBase + n];
        c = __builtin_amdgcn_wmma_f32_16x16x4_f32(false, a, false, b,
                                                  (short)0, c, false, false);
    }

    // D (16x16 f32): VGPR r -> row r (lanes 0-15) / row r+8 (lanes 16-31), col n.
    const int rOff = (lane >> 4) * 8;
    const float bias = isK ? bh[colBase + n] : 0.0f;
    for (int r = 0; r < 8; ++r) {
        out[(size_t)(rowBase + r + rOff) * UDIM + colBase + n] = c[r] + bias;
    }
}

// ---------------------------------------------------------------------------
// Kernel 2: per (b, 16-query tile):
//   e[ti][j] = Wa . tanh(q[t0+ti] + k[sLo+j])   (band: ti <= j < ti+64, s in [0,T))
//   row softmax over band -> A[16][80] in LDS
//   v[16][256] = A @ X[sLo..sLo+79][:]  via f32 WMMA (K=80, 20 steps, 16 col tiles)
// 128 threads = 4 waves; wave w handles F-column tiles 4w..4w+3.
// ---------------------------------------------------------------------------
__global__ __launch_bounds__(128) void attn_band_kernel(
    const float* __restrict__ x,     // [B*T, F]
    const float* __restrict__ qp,    // [B*T, U]
    const float* __restrict__ kp,    // [B*T, U]  (bh already folded in)
    const float* __restrict__ Wa,    // [U]
    float* __restrict__ v)           // [B*T, F]
{
    __shared__ float qs[16][UDIM + 1];     // +1 pad: conflict-free u access
    __shared__ float ks[SWIN][UDIM + 1];
    __shared__ float As[16][84];           // stride 84: conflict-free A-frag loads
    __shared__ float wa_s[UDIM];

    const int tid = threadIdx.x;
    const int b   = blockIdx.y;
    const int t0  = blockIdx.x * 16;
    const int sLo = t0 - BAND / 2;
    const size_t base = (size_t)b * TLEN;

    // Stage A: cooperative loads of q tile and clamped k window into LDS.
    for (int i = tid; i < 16 * UDIM; i += 128) {
        int r = i >> 5, u = i & 31;
        qs[r][u] = qp[(base + t0 + r) * UDIM + u];
    }
    for (int i = tid; i < SWIN * UDIM; i += 128) {
        int j = i >> 5, u = i & 31;
        int s = sLo + j;
        s = s < 0 ? 0 : (s > TLEN - 1 ? TLEN - 1 : s);
        ks[j][u] = kp[(base + s) * UDIM + u];
    }
    if (tid < UDIM) wa_s[tid] = Wa[tid];
    __syncthreads();

    // Stage B: band-limited logits e[16][80] (tanh on the trans pipe).
    for (int i = tid; i < 16 * SWIN; i += 128) {
        int ti = i / SWIN, j = i - ti * SWIN;
        int s = sLo + j;
        bool valid = (j >= ti) && (j < ti + BAND) && (s >= 0) && (s < TLEN);
        float e = -1e30f;
        if (valid) {
            float acc = 0.0f;
#pragma unroll
            for (int u = 0; u < UDIM; ++u)
                acc += wa_s[u] * tanhf(qs[ti][u] + ks[j][u]);
            e = acc;
        }
        As[ti][j] = e;
    }
    __syncthreads();

    // Stage C: row softmax (16 rows, one thread each; 80 entries/row).
    if (tid < 16) {
        const int ti = tid;
        float m = -1e30f;
        for (int j = 0; j < SWIN; ++j) m = fmaxf(m, As[ti][j]);
        float sum = 0.0f;
        for (int j = 0; j < SWIN; ++j) {
            float a = (As[ti][j] <= -1e29f) ? 0.0f : expf(As[ti][j] - m);
            As[ti][j] = a;
            sum += a;
        }
        const float inv = 1.0f / sum;
        for (int j = 0; j < SWIN; ++j) As[ti][j] *= inv;
    }
    __syncthreads();

    // Stage D: v[16x256] = A[16x80] @ X[80x256] with f32 WMMA.
    const int wave  = tid >> 5;
    const int lane  = tid & 31;
    const int mrow  = lane & 15;
    const int khalf = (lane >> 4) * 2;
    const int n     = lane & 15;
    const int rOff  = (lane >> 4) * 8;

    for (int nt = 0; nt < 4; ++nt) {
        const int colBase = (wave * 4 + nt) * 16;
        v8f c = {};
        for (int k0 = 0; k0 < SWIN; k0 += 4) {
            v2f a;
            a.x = As[mrow][k0 + khalf];
            a.y = As[mrow][k0 + khalf + 1];
            // B fragment from x, addresses clamped (weights there are exactly 0).
            int s0 = sLo + k0 + khalf;
            int s1 = s0 + 1;
            int sc0 = s0 < 0 ? 0 : (s0 > TLEN - 1 ? TLEN - 1 : s0);
            int sc1 = s1 < 0 ? 0 : (s1 > TLEN - 1 ? TLEN - 1 : s1);
            v2f bb;
            bb.x = x[(base + sc0) * FDIM + colBase + n];
            bb.y = x[(base + sc1) * FDIM + colBase + n];
            c = __builtin_amdgcn_wmma_f32_16x16x4_f32(false, a, false, bb,
                                                      (short)0, c, false, false);
        }
        for (int r = 0; r < 8; ++r) {
            v[(base + t0 + r + rOff) * FDIM + colBase + n] = c[r];
        }
    }
}

// ---------------------------------------------------------------------------
extern "C" void kernel_launch(void* const* d_in, const int* in_sizes, int n_in,
                              void* d_out, int out_size, void* d_ws, size_t ws_size,
                              hipStream_t stream) {
    (void)in_sizes; (void)n_in; (void)out_size; (void)ws_size;
    const float* x  = (const float*)d_in[0];   // [4,1024,256]
    const float* Wt = (const float*)d_in[1];   // [256,32]
    const float* Wx = (const float*)d_in[2];   // [256,32]
    const float* bh = (const float*)d_in[3];   // [32]
    const float* Wa = (const float*)d_in[4];   // [32,1]
    // d_in[5] = ba (constant shift of all logits -> cancels in softmax)

    float* qws = (float*)d_ws;                         // [4096,32] = 512 KB
    float* kws = qws + (size_t)BATCH * TLEN * UDIM;    // [4096,32] = 512 KB
    float* out = (float*)d_out;                        // [4,1024,256]

    dim3 g1((BATCH * TLEN) / 16, 4);
    proj_wmma_kernel<<<g1, 32, 0, stream>>>(x, Wt, Wx, bh, qws, kws);

    dim3 g2(TLEN / 16, BATCH);
    attn_band_kernel<<<g2, 128, 0, stream>>>(x, qws, kws, Wa, out);
}